// PredictionLoss_L2_sceneLoss_91130616087309
// MI455X (gfx1250) — compile-verified
//
#include <hip/hip_runtime.h>

// Problem constants from the reference
#define Bdim 64
#define Tdim 30
#define Sdim 4
#define Pdim 20000
#define Hdim 2160
#define Wdim 3840
#define SCENE_SCALE 10.0f
#define GAMA 5.0f

#define NTILES (Pdim / 16)   // 1250 road-point tiles of 16
#define WAVES 8
#define BLK 256

typedef __attribute__((ext_vector_type(2))) float v2f;
typedef __attribute__((ext_vector_type(8))) float v8f;

__global__ __launch_bounds__(BLK)
void scene_loss_kernel(const float* __restrict__ inputs,        // (B,T,2)
                       const float* __restrict__ targets,       // (B,T,2)
                       const float* __restrict__ start,         // (B,2)
                       const unsigned char* __restrict__ scene, // (S,H,W) bool
                       const int* __restrict__ road_points,     // (S,P,2)
                       const int* __restrict__ file_name,       // (B,)
                       float* __restrict__ partial)             // (B*2,)
{
    __shared__ float qx[16], qy[16], q2s[16], pxs[16], pys[16];
    __shared__ int   dxs[16], dys[16];
    __shared__ float lmin[WAVES][16][16];
    __shared__ int   lidx[WAVES][16][16];
    __shared__ float red[BLK];

    const int tid   = threadIdx.x;
    const int lane  = tid & 31;
    const int wave  = tid >> 5;
    const int b     = blockIdx.x;
    const int ttile = blockIdx.y;
    const int fn    = file_name[b];
    const int* rp   = road_points + (size_t)fn * (Pdim * 2);

    // --- per-tile query setup (16 queries) ---
    if (tid < 16) {
        const int m = tid;
        const int t = ttile * 16 + m;
        float fx = 0.f, fy = 0.f, gx = 0.f, gy = 0.f;
        int dx = 0, dy = 0;
        if (t < Tdim) {
            const float sx = start[b * 2 + 0], sy = start[b * 2 + 1];
            fx = inputs[((size_t)b * Tdim + t) * 2 + 0] + sx;   // unclipped position
            fy = inputs[((size_t)b * Tdim + t) * 2 + 1] + sy;
            dx = (int)fx; dx = dx < 0 ? 0 : (dx > Hdim - 1 ? Hdim - 1 : dx);
            dy = (int)fy; dy = dy < 0 ? 0 : (dy > Wdim - 1 ? Wdim - 1 : dy);
            gx = (float)dx; gy = (float)dy;                      // clipped int -> float query
        }
        qx[m] = gx; qy[m] = gy; q2s[m] = gx * gx + gy * gy;
        pxs[m] = fx; pys[m] = fy; dxs[m] = dx; dys[m] = dy;
    }
    __syncthreads();

    // --- A matrix (16x4 f32): lanes 0-15 hold K=0,1 ; lanes 16-31 hold K=2,3 (zero pad) ---
    const int  col = lane & 15;
    const bool lo  = lane < 16;
    v2f amat;
    amat.x = lo ? qx[col] : 0.0f;
    amat.y = lo ? qy[col] : 0.0f;

    float q2v[8];
#pragma unroll
    for (int j = 0; j < 8; ++j) q2v[j] = q2s[j + (lo ? 0 : 8)];

    float mind[8];
    int   mindi[8];
#pragma unroll
    for (int j = 0; j < 8; ++j) { mind[j] = 3.4e38f; mindi[j] = 0; }

    // --- main loop: each wave strides over road-point tiles ---
    for (int pt = wave; pt < NTILES; pt += WAVES) {
        const int p0   = pt * 16;
        const int ridx = p0 + col;
        __builtin_prefetch(rp + 2 * (ridx + 16 * WAVES), 0, 1); // global_prefetch_b8
        const float rx = (float)rp[2 * ridx + 0];
        const float ry = (float)rp[2 * ridx + 1];
        const float r2 = rx * rx + ry * ry;

        v2f bmat;                       // B matrix (4x16): K=0,1 in lanes 0-15, zero pad above
        bmat.x = lo ? rx : 0.0f;
        bmat.y = lo ? ry : 0.0f;

        v8f cz = {0.f, 0.f, 0.f, 0.f, 0.f, 0.f, 0.f, 0.f};
        // D[M][N] = sum_k A[M][k]*B[k][N]  (16x16 dot products in one instruction)
        v8f dm = __builtin_amdgcn_wmma_f32_16x16x4_f32(
            /*neg_a=*/false, amat, /*neg_b=*/false, bmat,
            /*c_mod=*/(short)0, cz, /*reuse_a=*/false, /*reuse_b=*/false);

#pragma unroll
        for (int j = 0; j < 8; ++j) {
            const float d2 = q2v[j] + r2 - 2.0f * dm[j];
            const bool better = d2 < mind[j];
            mind[j]  = better ? d2   : mind[j];
            mindi[j] = better ? ridx : mindi[j];
        }
    }

    // --- stash per-lane argmin candidates; D layout: M = j + 8*(lane>=16), N = lane&15 ---
#pragma unroll
    for (int j = 0; j < 8; ++j) {
        const int M = j + (lo ? 0 : 8);
        lmin[wave][M][col] = mind[j];
        lidx[wave][M][col] = mindi[j];
    }
    __syncthreads();

    // --- final argmin per query + scene loss term ---
    float contrib = 0.0f;
    if (tid < 16) {
        const int m = tid, t = ttile * 16 + m;
        if (t < Tdim) {
            float best = 3.4e38f; int besti = 0x7fffffff;
            for (int w = 0; w < WAVES; ++w)
                for (int c2 = 0; c2 < 16; ++c2) {
                    const float v = lmin[w][m][c2];
                    const int   i = lidx[w][m][c2];
                    if (v < best || (v == best && i < besti)) { best = v; besti = i; }
                }
            const float nx = (float)rp[2 * besti + 0];
            const float ny = (float)rp[2 * besti + 1];
            const float per = (fabsf(pxs[m] - nx) + fabsf(pys[m] - ny)) * SCENE_SCALE;
            const size_t off = (size_t)fn * Hdim * Wdim + (size_t)dxs[m] * Wdim + dys[m];
            contrib = scene[off] ? per : 0.0f;
        }
    }

    // --- L1 term for this (b, ttile): 16 t's x 2 comps handled by threads 0..31 ---
    float l1 = 0.0f;
    if (tid < 32) {
        const int t = ttile * 16 + (tid >> 1), c = tid & 1;
        if (t < Tdim) {
            const float a  = inputs[((size_t)b * Tdim + t) * 2 + c];
            const float bb = targets[((size_t)b * Tdim + t) * 2 + c];
            l1 = fabsf(a - bb);
        }
    }

    red[tid] = l1 + GAMA * contrib;
    __syncthreads();
    for (int s = BLK / 2; s > 0; s >>= 1) {
        if (tid < s) red[tid] += red[tid + s];
        __syncthreads();
    }
    if (tid == 0) partial[blockIdx.x * gridDim.y + blockIdx.y] = red[0];
}

__global__ void final_reduce_kernel(const float* __restrict__ partial,
                                    float* __restrict__ out, int n)
{
    if (blockIdx.x == 0 && threadIdx.x == 0) {
        float acc = 0.0f;
        for (int i = 0; i < n; ++i) acc += partial[i];
        out[0] = acc;
    }
}

extern "C" void kernel_launch(void* const* d_in, const int* in_sizes, int n_in,
                              void* d_out, int out_size, void* d_ws, size_t ws_size,
                              hipStream_t stream) {
    const float*         inputs   = (const float*)d_in[0];        // (B,T,2) f32
    const float*         targets  = (const float*)d_in[1];        // (B,T,2) f32
    const float*         start    = (const float*)d_in[2];        // (B,2)   f32
    const unsigned char* scene    = (const unsigned char*)d_in[3];// (S,H,W) bool
    const int*           roadpts  = (const int*)d_in[4];          // (S,P,2) i32
    const int*           filename = (const int*)d_in[5];          // (B,)    i32

    float* partial = (float*)d_ws;                                // 128 floats
    float* out     = (float*)d_out;

    dim3 grid(Bdim, 2);   // (batch, T-tile)
    scene_loss_kernel<<<grid, BLK, 0, stream>>>(inputs, targets, start, scene,
                                                roadpts, filename, partial);
    final_reduce_kernel<<<1, 32, 0, stream>>>(partial, out, Bdim * 2);
}